// HGCN_11158325035511
// MI455X (gfx1250) — compile-verified
//
#include <hip/hip_runtime.h>
#include <hip/hip_bf16.h>

// ---------------------------------------------------------------------------
// CDNA5 (gfx1250) HGCN pipeline.
// Dense parts use V_WMMA_F32_16X16X4_F32 (wave32, D = A[16x4] * B[4x16] + C).
// VGPR layouts per CDNA5 ISA 7.12.2:
//   A (16x4 f32):  lane L, vgpr j -> A[M = L%16, K = 2*(L/16) + j]
//   B (4x16 f32):  lane L, vgpr j -> B[K = j + 2*(L/16), N = L%16]
//   C/D (16x16):   lane L, vgpr i -> D[M = i + 8*(L/16), N = L%16]
// Scatter-add uses the CDNA5 hardware f32 atomic (global_atomic_add_f32,
// device scope, no-return form -> tracked by STOREcnt; S_ENDPGM implies
// wait-idle so completion is guaranteed before wave exit).
// ---------------------------------------------------------------------------

typedef float v2f __attribute__((ext_vector_type(2)));
typedef float v8f __attribute__((ext_vector_type(8)));

__device__ __forceinline__ v8f wmma_f32x4(v2f a, v2f b, v8f c) {
  // 8 args: (neg_a, A, neg_b, B, c_mod, C, reuse_a, reuse_b)
  return __builtin_amdgcn_wmma_f32_16x16x4_f32(
      false, a, false, b, (short)0, c, false, false);
}

__device__ __forceinline__ void atomic_add_f32_global(float* p, float v) {
  // Hardware FP32 atomic RMW at L2 (CDNA5 ISA 15.18.3 op 86), no return value.
  asm volatile("global_atomic_add_f32 %0, %1, off scope:SCOPE_DEV"
               :
               : "v"(p), "v"(v)
               : "memory");
}

// out[nrows x 64] = A[nrows x K] @ W[K x 64]  (+ bias)  (* (Ef@WeG + beG) if GATED)
// One wave computes a 32-row x 64-col strip: two 16-row A fragments share each
// B fragment (8 WMMAs per 10 loads per k-step).
template <int K, bool BIAS, bool GATED>
__global__ __launch_bounds__(256) void gemm64_kernel(
    const float* __restrict__ A, const float* __restrict__ W,
    const float* __restrict__ bias,
    const float* __restrict__ Ef,    // [nrows x 16] edge-gate features
    const float* __restrict__ WeG,   // [16 x 64]
    const float* __restrict__ beG,   // [64]
    float* __restrict__ out, int nstrips) {
  const int lane  = threadIdx.x & 31;
  const int strip = blockIdx.x * (blockDim.x >> 5) + (threadIdx.x >> 5);
  if (strip >= nstrips) return;  // wave-uniform: EXEC stays all-ones for WMMA
  const int lm   = lane & 15;
  const int lh   = lane >> 4;
  const int row0 = strip << 5;   // 32 rows per wave

  v8f acc0[4] = {};
  v8f acc1[4] = {};
  {
    const float* arow0 = A + (size_t)(row0 + lm) * K + 2 * lh;
    const float* arow1 = arow0 + (size_t)16 * K;
#pragma unroll
    for (int k = 0; k < K; k += 4) {
      v2f a0 = *(const v2f*)(arow0 + k);
      v2f a1 = *(const v2f*)(arow1 + k);
      const float* wk = W + (size_t)(k + 2 * lh) * 64 + lm;
#pragma unroll
      for (int t = 0; t < 4; ++t) {
        v2f b;
        b.x = wk[t * 16];
        b.y = wk[64 + t * 16];
        acc0[t] = wmma_f32x4(a0, b, acc0[t]);
        acc1[t] = wmma_f32x4(a1, b, acc1[t]);
      }
    }
  }

  v8f gacc0[4] = {};
  v8f gacc1[4] = {};
  if (GATED) {  // fused edge gate: (Ef @ WeG + beG), K = 16
    const float* erow0 = Ef + (size_t)(row0 + lm) * 16 + 2 * lh;
    const float* erow1 = erow0 + (size_t)16 * 16;
#pragma unroll
    for (int k = 0; k < 16; k += 4) {
      v2f a0 = *(const v2f*)(erow0 + k);
      v2f a1 = *(const v2f*)(erow1 + k);
      const float* wk = WeG + (size_t)(k + 2 * lh) * 64 + lm;
#pragma unroll
      for (int t = 0; t < 4; ++t) {
        v2f b;
        b.x = wk[t * 16];
        b.y = wk[64 + t * 16];
        gacc0[t] = wmma_f32x4(a0, b, gacc0[t]);
        gacc1[t] = wmma_f32x4(a1, b, gacc1[t]);
      }
    }
  }

#pragma unroll
  for (int t = 0; t < 4; ++t) {
    const int n = t * 16 + lm;
    const float bv = BIAS ? bias[n] : 0.0f;
    const float gb = GATED ? beG[n] : 0.0f;
#pragma unroll
    for (int i = 0; i < 8; ++i) {
      const int m0 = row0 + i + 8 * lh;
      float v0 = acc0[t][i] + bv;
      float v1 = acc1[t][i] + bv;
      if (GATED) {
        v0 *= (gacc0[t][i] + gb);
        v1 *= (gacc1[t][i] + gb);
      }
      out[(size_t)m0 * 64 + n]        = v0;
      out[(size_t)(m0 + 16) * 64 + n] = v1;
    }
  }
}

// Z[dst] += val * T[src]  (segment_sum). 16 threads per edge, float4 each.
// Atomics land on 25.6MB Z buffers -> resident in 192MB L2.
__global__ __launch_bounds__(256) void scatter_kernel(
    const float* __restrict__ T, const float* __restrict__ val,
    const int* __restrict__ src, const int* __restrict__ dst,
    float* __restrict__ Z, int nedges) {
  const int gid = blockIdx.x * blockDim.x + threadIdx.x;
  const int e = gid >> 4;
  if (e >= nedges) return;
  const int q = (gid & 15) << 2;
  const float v = val[e];
  const float4 t = *(const float4*)(T + (size_t)src[e] * 64 + q);
  float* zp = Z + (size_t)dst[e] * 64 + q;
  atomic_add_f32_global(zp + 0, v * t.x);
  atomic_add_f32_global(zp + 1, v * t.y);
  atomic_add_f32_global(zp + 2, v * t.z);
  atomic_add_f32_global(zp + 3, v * t.w);
}

__global__ __launch_bounds__(256) void relu_kernel(
    const float4* __restrict__ Z, float4* __restrict__ Hout, int n4) {
  const int i = blockIdx.x * blockDim.x + threadIdx.x;
  if (i >= n4) return;
  float4 z = Z[i];
  z.x = fmaxf(z.x, 0.0f);
  z.y = fmaxf(z.y, 0.0f);
  z.z = fmaxf(z.z, 0.0f);
  z.w = fmaxf(z.w, 0.0f);
  Hout[i] = z;
}

extern "C" void kernel_launch(void* const* d_in, const int* in_sizes, int n_in,
                              void* d_out, int out_size, void* d_ws, size_t ws_size,
                              hipStream_t stream) {
  constexpr int N = 100000, E = 800000, H = 64;
  constexpr size_t NH = (size_t)N * H;

  const float* feat_a = (const float*)d_in[0];
  const float* Wp_a   = (const float*)d_in[1];
  const float* bp_a   = (const float*)d_in[2];
  const float* feat_b = (const float*)d_in[3];
  const float* Wp_b   = (const float*)d_in[4];
  const float* bp_b   = (const float*)d_in[5];

  // relations in dict order: 0=aa, 1=ab, 2=ba, 3=bb  (src type, dst type)
  const float* efeat[4]; const float* We[4]; const float* be[4];
  const float* val[4]; const int* src[4]; const int* dst[4];
  for (int r = 0; r < 4; ++r) {
    const int b = 6 + 6 * r;
    efeat[r] = (const float*)d_in[b + 0];
    We[r]    = (const float*)d_in[b + 1];
    be[r]    = (const float*)d_in[b + 2];
    val[r]   = (const float*)d_in[b + 3];
    src[r]   = (const int*)d_in[b + 4];
    dst[r]   = (const int*)d_in[b + 5];
  }
  const float* Ws[2][2]; const float* Wr[2][4];
  for (int l = 0; l < 2; ++l) {
    const int b = 30 + 6 * l;
    Ws[l][0] = (const float*)d_in[b + 0];  // Ws_a_l
    Ws[l][1] = (const float*)d_in[b + 1];  // Ws_b_l
    for (int r = 0; r < 4; ++r) Wr[l][r] = (const float*)d_in[b + 2 + r];
  }
  const float* Wout = (const float*)d_in[42];
  float* out = (float*)d_out;

  // workspace: Ha, Hb, Za, Zb, T  (5 x 25.6 MB)
  float* Ha = (float*)d_ws;
  float* Hb = Ha + NH;
  float* Za = Hb + NH;
  float* Zb = Za + NH;
  float* T  = Zb + NH;

  const int nstrips    = N / 32;              // 3125 strips of 32 rows
  const int gemmBlocks = (nstrips + 7) / 8;   // 8 waves (strips) per block
  const int scatBlocks = (E * 16) / 256;      // 50000
  const int n4         = (int)(NH / 4);
  const int reluBlocks = (n4 + 255) / 256;

  // input projections: Ha = feat_a @ Wp_a + bp_a ; same for b (K = 128)
  gemm64_kernel<128, true, false><<<gemmBlocks, 256, 0, stream>>>(
      feat_a, Wp_a, bp_a, nullptr, nullptr, nullptr, Ha, nstrips);
  gemm64_kernel<128, true, false><<<gemmBlocks, 256, 0, stream>>>(
      feat_b, Wp_b, bp_b, nullptr, nullptr, nullptr, Hb, nstrips);

  for (int l = 0; l < 2; ++l) {
    // self terms initialize Z (no memset needed)
    gemm64_kernel<64, false, false><<<gemmBlocks, 256, 0, stream>>>(
        Ha, Ws[l][0], nullptr, nullptr, nullptr, nullptr, Za, nstrips);
    gemm64_kernel<64, false, false><<<gemmBlocks, 256, 0, stream>>>(
        Hb, Ws[l][1], nullptr, nullptr, nullptr, nullptr, Zb, nstrips);
    for (int r = 0; r < 4; ++r) {
      const float* Hsrc = (r < 2) ? Ha : Hb;          // aa,ab src=a ; ba,bb src=b
      float* Zdst = ((r & 1) == 0) ? Za : Zb;         // aa,ba dst=a ; ab,bb dst=b
      // T = (Hsrc @ Wr) * (efeat @ We + be), gate fused via 2nd WMMA accumulator
      gemm64_kernel<64, false, true><<<gemmBlocks, 256, 0, stream>>>(
          Hsrc, Wr[l][r], nullptr, efeat[r], We[r], be[r], T, nstrips);
      scatter_kernel<<<scatBlocks, 256, 0, stream>>>(
          T, val[r], src[r], dst[r], Zdst, E);
    }
    relu_kernel<<<reluBlocks, 256, 0, stream>>>((const float4*)Za, (float4*)Ha, n4);
    relu_kernel<<<reluBlocks, 256, 0, stream>>>((const float4*)Zb, (float4*)Hb, n4);
  }

  // outputs: [2, N, 64] stacked flat
  gemm64_kernel<64, false, false><<<gemmBlocks, 256, 0, stream>>>(
      Ha, Wout, nullptr, nullptr, nullptr, nullptr, out, nstrips);
  gemm64_kernel<64, false, false><<<gemmBlocks, 256, 0, stream>>>(
      Hb, Wout, nullptr, nullptr, nullptr, nullptr, out + NH, nstrips);
}